// NeighboringLoss_70300024701835
// MI455X (gfx1250) — compile-verified
//
#include <hip/hip_runtime.h>
#include <hip/hip_bf16.h>

typedef __attribute__((ext_vector_type(2))) float v2f;
typedef __attribute__((ext_vector_type(8))) float v8f;

#define N_INST 256
#define LSEP 300.0f
#define LMEAN 300.0f

// workspace layout (floats), per image
#define OFF_CNT    0        // 256
#define OFF_SUM    256      // 768
#define OFF_MEAN   1024     // 768
#define OFF_M2     1792     // 256
#define OFF_INTRA  2048     // 256
#define OFF_SEP    2304     // 256
#define OFF_SEPCNT 2560     // 256
#define OFF_ROWANY 2816     // 256
#define OFF_ADJ    3072     // 2048 uint32 (256 rows x 8 words of adj2 bits)
#define IMG_STRIDE 5120

// ---------------- counts & sums (segment sum via LDS + global atomics) ----------------
__global__ __launch_bounds__(256) void nl_count(const float* __restrict__ pred,
                                                const float* __restrict__ tgt,
                                                float* __restrict__ ws, int P) {
  int b = blockIdx.y;
  const float* p = pred + (size_t)b * 3 * P;
  const float* t = tgt + (size_t)b * 3 * P;
  float* cnt = ws + (size_t)b * IMG_STRIDE + OFF_CNT;
  float* sums = ws + (size_t)b * IMG_STRIDE + OFF_SUM;
  __shared__ float lc[N_INST], ls0[N_INST], ls1[N_INST], ls2[N_INST];
  for (int i = threadIdx.x; i < N_INST; i += blockDim.x) { lc[i] = 0.f; ls0[i] = 0.f; ls1[i] = 0.f; ls2[i] = 0.f; }
  __syncthreads();
  for (int i = blockIdx.x * blockDim.x + threadIdx.x; i < P; i += gridDim.x * blockDim.x) {
    float t0 = t[i], t1 = t[P + i], t2 = t[2 * P + i];
    int l = (int)(t0 * 65536.0f + t1 * 256.0f + t2);
    l = min(max(l, 0), N_INST - 1);
    atomicAdd(&lc[l], 1.0f);
    atomicAdd(&ls0[l], p[i]);
    atomicAdd(&ls1[l], p[P + i]);
    atomicAdd(&ls2[l], p[2 * P + i]);
  }
  __syncthreads();
  for (int i = threadIdx.x; i < N_INST; i += blockDim.x) {
    if (lc[i] != 0.f) {
      atomicAdd(&cnt[i], lc[i]);
      atomicAdd(&sums[3 * i + 0], ls0[i]);
      atomicAdd(&sums[3 * i + 1], ls1[i]);
      atomicAdd(&sums[3 * i + 2], ls2[i]);
    }
  }
}

// ---------------- means, m2, row_any, adj2 bitmask, sep_cnt ----------------
__global__ __launch_bounds__(256) void nl_means(const unsigned char* __restrict__ adj,
                                                const unsigned char* __restrict__ nobg,
                                                float* __restrict__ ws) {
  int b = blockIdx.x;
  float* base = ws + (size_t)b * IMG_STRIDE;
  int k = threadIdx.x;
  __shared__ float scnt[N_INST];
  scnt[k] = base[OFF_CNT + k];
  __syncthreads();
  float c = scnt[k];
  float inv = 1.0f / fmaxf(c, 1.0f);
  float mx = base[OFF_SUM + 3 * k + 0] * inv;
  float my = base[OFF_SUM + 3 * k + 1] * inv;
  float mz = base[OFF_SUM + 3 * k + 2] * inv;
  base[OFF_MEAN + 3 * k + 0] = mx;
  base[OFF_MEAN + 3 * k + 1] = my;
  base[OFF_MEAN + 3 * k + 2] = mz;
  base[OFF_M2 + k] = mx * mx + my * my + mz * mz;

  const unsigned char* row = adj + (size_t)k * N_INST;
  int any = 0;
  for (int j = 0; j < N_INST; j++) any |= (row[j] != 0);
  bool nb = (nobg[b] != 0);
  unsigned wbits[8] = {0, 0, 0, 0, 0, 0, 0, 0};
  float sc = 0.f;
  for (int j = 0; j < N_INST; j++) {
    int a2 = (j == 0) ? ((nb && any) ? 1 : (row[0] != 0)) : (row[j] != 0);
    if (a2) {
      wbits[j >> 5] |= (1u << (j & 31));
      sc += scnt[j];
    }
  }
  unsigned* gadj = (unsigned*)(base + OFF_ADJ);
  #pragma unroll
  for (int w = 0; w < 8; w++) gadj[k * 8 + w] = wbits[w];
  base[OFF_ROWANY + k] = any ? 1.0f : 0.0f;
  base[OFF_SEPCNT + k] = sc;
}

// ---------------- intra: Huber vs instance mean ----------------
__global__ __launch_bounds__(256) void nl_intra(const float* __restrict__ pred,
                                                const float* __restrict__ tgt,
                                                float* __restrict__ ws, int P) {
  int b = blockIdx.y;
  float* base = ws + (size_t)b * IMG_STRIDE;
  const float* p = pred + (size_t)b * 3 * P;
  const float* t = tgt + (size_t)b * 3 * P;
  __shared__ float sm[3 * N_INST];
  __shared__ float lh[N_INST];
  for (int i = threadIdx.x; i < 3 * N_INST; i += blockDim.x) sm[i] = base[OFF_MEAN + i];
  for (int i = threadIdx.x; i < N_INST; i += blockDim.x) lh[i] = 0.f;
  __syncthreads();
  for (int i = blockIdx.x * blockDim.x + threadIdx.x; i < P; i += gridDim.x * blockDim.x) {
    float t0 = t[i], t1 = t[P + i], t2 = t[2 * P + i];
    int l = (int)(t0 * 65536.0f + t1 * 256.0f + t2);
    l = min(max(l, 0), N_INST - 1);
    float mx = (l == 0) ? 0.f : sm[3 * l + 0];
    float my = (l == 0) ? 0.f : sm[3 * l + 1];
    float mz = (l == 0) ? 0.f : sm[3 * l + 2];
    float dx = p[i] - mx, dy = p[P + i] - my, dz = p[2 * P + i] - mz;
    float ax = fabsf(dx), ay = fabsf(dy), az = fabsf(dz);
    float h = ((ax < 1.f) ? 0.5f * dx * dx : ax - 0.5f)
            + ((ay < 1.f) ? 0.5f * dy * dy : ay - 0.5f)
            + ((az < 1.f) ? 0.5f * dz * dz : az - 0.5f);
    atomicAdd(&lh[l], h);
  }
  __syncthreads();
  for (int i = threadIdx.x; i < N_INST; i += blockDim.x)
    if (lh[i] != 0.f) atomicAdd(&base[OFF_INTRA + i], lh[i]);
}

// ---------------- inter: masked RBF of every pixel vs every mean, via WMMA f32 16x16x4 ----
// partial = A(16x4) x B(4x16) + 0, with A row = [px,py,pz,p^2], B col = [-2mx,-2my,-2mz,1]
// => partial[i][j] = p_i^2 - 2 p_i.m_j.  Then dist2+1 = partial + (1 + m_j^2), folded into
// the per-tile scalar m2p1, so C is the inline-zero operand (no accumulator init movs).
__global__ __launch_bounds__(256) void nl_inter(const float* __restrict__ pred,
                                                const float* __restrict__ tgt,
                                                float* __restrict__ ws, int P) {
  int b = blockIdx.y;
  float* base = ws + (size_t)b * IMG_STRIDE;
  const float* p = pred + (size_t)b * 3 * P;
  const float* t = tgt + (size_t)b * 3 * P;
  int tid = threadIdx.x;
  __shared__ float sm[3 * N_INST];
  __shared__ float sm2[N_INST];
  __shared__ unsigned sadj[8 * N_INST];
  __shared__ float ssep[N_INST];
  for (int i = tid; i < 3 * N_INST; i += 256) sm[i] = base[OFF_MEAN + i];
  for (int i = tid; i < N_INST; i += 256) { sm2[i] = base[OFF_M2 + i]; ssep[i] = 0.f; }
  const unsigned* gadj = (const unsigned*)(base + OFF_ADJ);
  for (int i = tid; i < 8 * N_INST; i += 256) sadj[i] = gadj[i];
  __syncthreads();

  int lane = tid & 31;
  bool hi = lane >= 16;
  int col = lane & 15;

  // per-tile B matrices and (1 + m^2) broadcast values (loop-invariant)
  v2f bm[16];
  float m2p1[16];
  #pragma unroll
  for (int tI = 0; tI < 16; tI++) {
    int j = tI * 16 + col;
    float mx = sm[3 * j + 0], my = sm[3 * j + 1], mz = sm[3 * j + 2];
    bm[tI].x = hi ? (-2.0f * mz) : (-2.0f * mx);   // K=2 (hi) / K=0 (lo)
    bm[tI].y = hi ? 1.0f : (-2.0f * my);           // K=3 (hi) / K=1 (lo)
    m2p1[tI] = 1.0f + sm2[j];
  }
  float acc[16];
  #pragma unroll
  for (int tI = 0; tI < 16; tI++) acc[tI] = 0.f;

  int gw = (blockIdx.x * 256 + tid) >> 5;            // global wave id
  int nwaves = gridDim.x * 8;
  for (int pb = gw * 16; pb < P; pb += nwaves * 16) {  // uniform per wave; P % 16 == 0
    int i = pb + col;                                  // lanes L and L+16 load the same pixel
    float px = p[i], py = p[P + i], pz = p[2 * P + i];
    float p2 = px * px + py * py + pz * pz;
    float t0 = t[i], t1 = t[P + i], t2 = t[2 * P + i];
    int myl = (int)(t0 * 65536.0f + t1 * 256.0f + t2);
    myl = min(max(myl, 0), N_INST - 1);
    v2f a;
    a.x = hi ? pz : px;                                // A: K=2 / K=0
    a.y = hi ? p2 : py;                                // A: K=3 / K=1
    int lbl[8];
    #pragma unroll
    for (int r = 0; r < 8; r++)                        // label of pixel M = r + 8*hi
      lbl[r] = __shfl(myl, r + (hi ? 8 : 0), 32);
    #pragma unroll
    for (int tI = 0; tI < 16; tI++) {
      int k = tI * 16 + col;                           // instance (D column) for this lane
      v8f c = {};                                      // inline-zero C operand
      v8f d = __builtin_amdgcn_wmma_f32_16x16x4_f32(
          false, a, false, bm[tI], (short)0, c, false, false);
      float s = 0.f;
      #pragma unroll
      for (int r = 0; r < 8; r++) {
        int lb = lbl[r];
        unsigned w = sadj[k * 8 + (lb >> 5)];
        float on = (float)((w >> (lb & 31)) & 1u);
        // 300/(1+dist2) with hardware reciprocal (1 TRANS op, co-executes with VALU)
        s += on * (LSEP * __builtin_amdgcn_rcpf(d[r] + m2p1[tI]));
      }
      acc[tI] += s;
    }
  }
  // lanes L and L+16 hold row-halves of the same column: combine, then flush
  #pragma unroll
  for (int tI = 0; tI < 16; tI++) {
    float v = acc[tI] + __shfl_xor(acc[tI], 16, 32);
    if (!hi) atomicAdd(&ssep[tI * 16 + col], v);
  }
  __syncthreads();
  for (int i = tid; i < N_INST; i += 256)
    if (ssep[i] != 0.f) atomicAdd(&base[OFF_SEP + i], ssep[i]);
}

// ---------------- per-image finalize: mean-pair penalty + combine ----------------
__device__ float nl_block_sum(float v, float* buf, int tid) {
  buf[tid] = v;
  __syncthreads();
  for (int s = 128; s > 0; s >>= 1) {
    if (tid < s) buf[tid] += buf[tid + s];
    __syncthreads();
  }
  float r = buf[0];
  __syncthreads();
  return r;
}

__global__ __launch_bounds__(256) void nl_final(const unsigned char* __restrict__ adj,
                                                const unsigned char* __restrict__ nobg,
                                                float* __restrict__ ws, int B) {
  int b = blockIdx.x;
  float* base = ws + (size_t)b * IMG_STRIDE;
  int k = threadIdx.x;
  __shared__ float smp[3 * N_INST];
  __shared__ unsigned char spres[N_INST];
  __shared__ float red[N_INST];
  float c = base[OFF_CNT + k];
  bool present = c > 0.f;
  smp[3 * k + 0] = (k == 0) ? 0.f : base[OFF_MEAN + 3 * k + 0];
  smp[3 * k + 1] = (k == 0) ? 0.f : base[OFF_MEAN + 3 * k + 1];
  smp[3 * k + 2] = (k == 0) ? 0.f : base[OFF_MEAN + 3 * k + 2];
  spres[k] = present ? 1 : 0;
  __syncthreads();

  // mean-level repulsion over upper-triangle pairs (original adj, mp[0]=0)
  float pen = 0.f, np = 0.f;
  if (present) {
    const unsigned char* row = adj + (size_t)k * N_INST;
    float ax = smp[3 * k + 0], ay = smp[3 * k + 1], az = smp[3 * k + 2];
    for (int j = k + 1; j < N_INST; j++) {
      if (spres[j] && row[j]) {
        float dx = ax - smp[3 * j + 0], dy = ay - smp[3 * j + 1], dz = az - smp[3 * j + 2];
        float dmm = dx * dx + dy * dy + dz * dz;
        pen += LMEAN / (dmm + 1.0f);
        np += 1.0f;
      }
    }
  }
  // intra term
  float ih = base[OFF_INTRA + k] / (3.0f * fmaxf(c, 1.0f));
  bool nb = (nobg[b] != 0);
  bool inc = present && ((k == 0) ? !nb : true);
  float intra_v = inc ? ih : 0.f;
  float ct_v = inc ? 1.f : 0.f;
  // inter term
  float sc = base[OFF_SEPCNT + k];
  float sep = base[OFF_SEP + k] / fmaxf(sc, 1.0f);
  float w = 10.0f / sqrtf(fmaxf(c, 1.0f));
  bool rowAny = base[OFF_ROWANY + k] != 0.f;
  bool valid = present && rowAny && (k > 0) && (sc > 0.f);
  float inter_v = valid ? (w * sep) : 0.f;

  float penT = nl_block_sum(pen, red, k);
  float npT = nl_block_sum(np, red, k);
  float intraT = nl_block_sum(intra_v, red, k);
  float ctT = nl_block_sum(ct_v, red, k);
  float interT = nl_block_sum(inter_v, red, k);
  if (k == 0) {
    float mean_pen = (npT > 0.f) ? (penT / fmaxf(npT, 1.0f)) : 0.f;
    float ct = fmaxf(ctT, 1.0f);
    float* pi = ws + (size_t)B * IMG_STRIDE;
    pi[b] = (intraT + interT + mean_pen) / ct;
  }
}

__global__ void nl_mean(const float* __restrict__ ws, float* __restrict__ out, int B) {
  float s = 0.f;
  const float* pi = ws + (size_t)B * IMG_STRIDE;
  for (int b = 0; b < B; b++) s += pi[b];
  out[0] = s / (float)B;
}

extern "C" void kernel_launch(void* const* d_in, const int* in_sizes, int n_in,
                              void* d_out, int out_size, void* d_ws, size_t ws_size,
                              hipStream_t stream) {
  const float* pred = (const float*)d_in[0];
  const float* tgt = (const float*)d_in[1];
  const unsigned char* nobg = (const unsigned char*)d_in[2];
  const unsigned char* adj = (const unsigned char*)d_in[3];
  float* ws = (float*)d_ws;

  int B = in_sizes[2];                 // no_bg has one entry per image
  int P = in_sizes[0] / (3 * B);       // pixels per image

  size_t ws_bytes = ((size_t)B * IMG_STRIDE + B) * sizeof(float);
  hipMemsetAsync(d_ws, 0, ws_bytes, stream);

  dim3 gridP(256, B);
  nl_count<<<gridP, 256, 0, stream>>>(pred, tgt, ws, P);
  nl_means<<<B, 256, 0, stream>>>(adj, nobg, ws);
  nl_intra<<<gridP, 256, 0, stream>>>(pred, tgt, ws, P);
  nl_inter<<<gridP, 256, 0, stream>>>(pred, tgt, ws, P);
  nl_final<<<B, 256, 0, stream>>>(adj, nobg, ws, B);
  nl_mean<<<1, 1, 0, stream>>>(ws, (float*)d_out, B);
}